// GNNLayerAttention_49727131353586
// MI455X (gfx1250) — compile-verified
//
#include <hip/hip_runtime.h>
#include <hip/hip_bf16.h>
#include <math.h>

typedef __attribute__((ext_vector_type(2))) float v2f;
typedef __attribute__((ext_vector_type(8))) float v8f;

__device__ __forceinline__ float leaky(float x) { return x >= 0.0f ? x : 0.2f * x; }

// order-preserving float -> uint key (for atomic segment-max)
__device__ __forceinline__ unsigned f2key(float f) {
    unsigned u = __float_as_uint(f);
    return (u & 0x80000000u) ? ~u : (u | 0x80000000u);
}
__device__ __forceinline__ float key2f(unsigned k) {
    unsigned u = (k & 0x80000000u) ? (k & 0x7FFFFFFFu) : ~k;
    return __uint_as_float(u);
}

// ---------------------------------------------------------------------------
// Kernel 0: zero scratch (h_neigh + denom + maxkey, contiguous)
// ---------------------------------------------------------------------------
__global__ void gat_zero_kernel(unsigned* __restrict__ p, size_t nwords) {
    size_t i = (size_t)blockIdx.x * blockDim.x + threadIdx.x;
    size_t stride = (size_t)gridDim.x * blockDim.x;
    for (; i < nwords; i += stride) p[i] = 0u;
}

// ---------------------------------------------------------------------------
// Kernel 1: h_trans = X @ W^T + b   (N x 64) = (N x 64)(64 x 64)^T
// One 16x16 output tile per wave, 4 waves per block (block covers 16 rows x 64 cols).
// Uses V_WMMA_F32_16X16X4_F32, full f32 precision.
// A layout (16x4 f32): lanes 0-15 = rows M, K=kh..kh+1 in v[0..1]; lanes 16-31 same
// rows, K=kh+2..kh+3. B layout (4x16): lanes hold column N, same K split.
// ---------------------------------------------------------------------------
__global__ void __launch_bounds__(128)
gat_gemm_att(const float* __restrict__ X, const float* __restrict__ W,
             const float* __restrict__ b, float* __restrict__ Y, int N)
{
    const int lane = threadIdx.x & 31;
    const int wave = threadIdx.x >> 5;
    const int l15  = lane & 15;
    const int half = lane >> 4;            // 0 or 1
    const int m0   = blockIdx.x * 16;
    const int n0   = wave * 16;
    const int row  = m0 + l15;
    const int rr   = (row < N) ? row : (N - 1);   // clamp loads; stores guarded
    const int col  = n0 + l15;
    const int kh   = half * 2;

    v8f acc = {};
#pragma unroll
    for (int k = 0; k < 64; k += 4) {
        const float2 av = *(const float2*)(X + (size_t)rr * 64 + k + kh);
        const float2 bv = *(const float2*)(W + (size_t)col * 64 + k + kh);
        v2f a;  a[0] = av.x;  a[1] = av.y;
        v2f bb; bb[0] = bv.x; bb[1] = bv.y;
        acc = __builtin_amdgcn_wmma_f32_16x16x4_f32(false, a, false, bb,
                                                    (short)0, acc, false, false);
    }
    const float bias = b[col];
#pragma unroll
    for (int r = 0; r < 8; ++r) {
        const int orow = m0 + half * 8 + r;   // C/D layout: lanes>=16 hold M=8..15
        if (orow < N) Y[(size_t)orow * 64 + col] = acc[r] + bias;
    }
}

// ---------------------------------------------------------------------------
// Kernel 2: per-edge attention score + segment max over dst.
// One wave per edge; each lane handles a float2 slice of the 64-dim reduction.
// ---------------------------------------------------------------------------
__global__ void __launch_bounds__(256)
gat_edge_scores(const int* __restrict__ src, const int* __restrict__ dst,
                const float* __restrict__ h_trans, const float* __restrict__ e_r,
                float* __restrict__ e_ij, unsigned* __restrict__ maxkey, int E)
{
    const int lane   = threadIdx.x & 31;
    const int warp   = (int)((blockIdx.x * blockDim.x + threadIdx.x) >> 5);
    const int nwarps = (int)((gridDim.x * blockDim.x) >> 5);
    const float2* ht = (const float2*)h_trans;
    const float2* er = (const float2*)e_r;

    for (int e = warp; e < E; e += nwarps) {
        const int s = src[e];
        const int d = dst[e];
        const float2 hs = ht[(size_t)s * 32 + lane];
        const float2 hd = ht[(size_t)d * 32 + lane];
        const float2 ev = er[(size_t)e * 32 + lane];
        float p = hd.x * tanhf(hs.x + ev.x) + hd.y * tanhf(hs.y + ev.y);
#pragma unroll
        for (int off = 16; off > 0; off >>= 1) p += __shfl_xor(p, off, 32);
        if (lane == 0) {
            const float v = leaky(p * 0.125f);   // scaling = 1/sqrt(64)
            e_ij[e] = v;
            atomicMax(&maxkey[d], f2key(v));
        }
    }
}

// ---------------------------------------------------------------------------
// Kernel 3: exp_e = exp(e_ij - max[dst]); denom[dst] += exp_e  (segment sum)
// ---------------------------------------------------------------------------
__global__ void __launch_bounds__(256)
gat_edge_exp(const int* __restrict__ dst, const float* __restrict__ e_ij,
             const unsigned* __restrict__ maxkey, float* __restrict__ exp_e,
             float* __restrict__ denom, int E)
{
    int e = blockIdx.x * blockDim.x + threadIdx.x;
    int stride = gridDim.x * blockDim.x;
    for (; e < E; e += stride) {
        const int d = dst[e];
        const float x = __expf(e_ij[e] - key2f(maxkey[d]));
        exp_e[e] = x;
        atomicAdd(&denom[d], x);
    }
}

// ---------------------------------------------------------------------------
// Kernel 4: h_neigh[dst] += features[src] * alpha  (weighted scatter)
// One wave per edge; float2 per lane -> two f32 global atomic adds.
// ---------------------------------------------------------------------------
__global__ void __launch_bounds__(256)
gat_edge_scatter(const int* __restrict__ src, const int* __restrict__ dst,
                 const float* __restrict__ features, const float* __restrict__ exp_e,
                 const float* __restrict__ denom, float* __restrict__ h_neigh, int E)
{
    const int lane   = threadIdx.x & 31;
    const int warp   = (int)((blockIdx.x * blockDim.x + threadIdx.x) >> 5);
    const int nwarps = (int)((gridDim.x * blockDim.x) >> 5);
    const float2* fv = (const float2*)features;

    for (int e = warp; e < E; e += nwarps) {
        const int s = src[e];
        const int d = dst[e];
        const float a = exp_e[e] / (denom[d] + 1e-9f);
        const float2 f = fv[(size_t)s * 32 + lane];
        atomicAdd(&h_neigh[(size_t)d * 64 + lane * 2 + 0], f.x * a);
        atomicAdd(&h_neigh[(size_t)d * 64 + lane * 2 + 1], f.y * a);
    }
}

// ---------------------------------------------------------------------------
// Kernel 5: h_out = leaky((X+H) @ W1^T + b1 + (X*H) @ W2^T + b2)
// Same WMMA tiling as kernel 1, two chained f32 WMMA accumulations per k-step.
// ---------------------------------------------------------------------------
__global__ void __launch_bounds__(128)
gat_gemm_out(const float* __restrict__ X, const float* __restrict__ H,
             const float* __restrict__ W1, const float* __restrict__ b1,
             const float* __restrict__ W2, const float* __restrict__ b2,
             float* __restrict__ out, int N)
{
    const int lane = threadIdx.x & 31;
    const int wave = threadIdx.x >> 5;
    const int l15  = lane & 15;
    const int half = lane >> 4;
    const int m0   = blockIdx.x * 16;
    const int n0   = wave * 16;
    const int row  = m0 + l15;
    const int rr   = (row < N) ? row : (N - 1);
    const int col  = n0 + l15;
    const int kh   = half * 2;

    v8f acc = {};
#pragma unroll
    for (int k = 0; k < 64; k += 4) {
        const float2 xf = *(const float2*)(X + (size_t)rr * 64 + k + kh);
        const float2 hf = *(const float2*)(H + (size_t)rr * 64 + k + kh);
        const float2 w1 = *(const float2*)(W1 + (size_t)col * 64 + k + kh);
        const float2 w2 = *(const float2*)(W2 + (size_t)col * 64 + k + kh);
        v2f a1; a1[0] = xf.x + hf.x; a1[1] = xf.y + hf.y;   // sum_h
        v2f a2; a2[0] = xf.x * hf.x; a2[1] = xf.y * hf.y;   // prod_h
        v2f c1; c1[0] = w1.x; c1[1] = w1.y;
        v2f c2; c2[0] = w2.x; c2[1] = w2.y;
        acc = __builtin_amdgcn_wmma_f32_16x16x4_f32(false, a1, false, c1,
                                                    (short)0, acc, false, false);
        acc = __builtin_amdgcn_wmma_f32_16x16x4_f32(false, a2, false, c2,
                                                    (short)0, acc, false, false);
    }
    const float bias = b1[col] + b2[col];
#pragma unroll
    for (int r = 0; r < 8; ++r) {
        const int orow = m0 + half * 8 + r;
        if (orow < N) out[(size_t)orow * 64 + col] = leaky(acc[r] + bias);
    }
}

// ---------------------------------------------------------------------------
// Host-side launcher
// ---------------------------------------------------------------------------
extern "C" void kernel_launch(void* const* d_in, const int* in_sizes, int n_in,
                              void* d_out, int out_size, void* d_ws, size_t ws_size,
                              hipStream_t stream) {
    const int*   idx      = (const int*)d_in[0];     // (2, E)
    const float* features = (const float*)d_in[1];   // (N, 64)
    const float* e_r      = (const float*)d_in[2];   // (E, 64)
    const float* W1       = (const float*)d_in[4];
    const float* b1       = (const float*)d_in[5];
    const float* W2       = (const float*)d_in[6];
    const float* b2       = (const float*)d_in[7];
    const float* Wa       = (const float*)d_in[8];
    const float* ba       = (const float*)d_in[9];

    const int N = in_sizes[1] / 64;
    const int E = in_sizes[2] / 64;
    const int* src = idx;
    const int* dst = idx + E;

    // workspace layout (floats): h_trans | h_neigh | denom | maxkey | e_ij | exp_e
    float*    ws      = (float*)d_ws;
    float*    h_trans = ws;
    float*    h_neigh = ws + (size_t)N * 64;
    float*    denom   = ws + (size_t)N * 128;
    unsigned* maxkey  = (unsigned*)(ws + (size_t)N * 129);
    float*    e_ij    = ws + (size_t)N * 130;
    float*    exp_e   = e_ij + (size_t)E;

    // 0) zero h_neigh + denom + maxkey (contiguous N*66 words)
    {
        const size_t zwords = (size_t)N * 66;
        int blocks = (int)((zwords + 255) / 256);
        if (blocks > 65535) blocks = 65535;
        gat_zero_kernel<<<blocks, 256, 0, stream>>>((unsigned*)h_neigh, zwords);
    }

    // 1) h_trans = features @ Wa^T + ba  (WMMA f32)
    gat_gemm_att<<<(N + 15) / 16, 128, 0, stream>>>(features, Wa, ba, h_trans, N);

    // 2) edge scores + segment max
    gat_edge_scores<<<4096, 256, 0, stream>>>(src, dst, h_trans, e_r, e_ij, maxkey, E);

    // 3) exp + segment sum
    gat_edge_exp<<<(E + 255) / 256, 256, 0, stream>>>(dst, e_ij, maxkey, exp_e, denom, E);

    // 4) weighted scatter aggregation
    gat_edge_scatter<<<4096, 256, 0, stream>>>(src, dst, features, exp_e, denom, h_neigh, E);

    // 5) fused output GEMMs + leaky (WMMA f32)
    gat_gemm_out<<<(N + 15) / 16, 128, 0, stream>>>(features, h_neigh, W1, b1, W2, b2,
                                                    (float*)d_out, N);
}